// GraphAttentionEmbedding_89893665505357
// MI455X (gfx1250) — compile-verified
//
#include <hip/hip_runtime.h>
#include <hip/hip_bf16.h>

typedef __attribute__((ext_vector_type(2))) float v2f;
typedef __attribute__((ext_vector_type(8))) float v8f;

#define NND 100000
#define NED 500000
#define HEADS 8
#define IN_CH 128
#define OUT_CH 128
#define HEAD_DIM 16
#define MSG_DIM 64
#define TIME_DIM 64

#define WMMA_F32(A, B, C) \
    __builtin_amdgcn_wmma_f32_16x16x4_f32(false, (A), false, (B), (short)0, (C), false, false)

// ---------------------------------------------------------------------------
// Kernel 1: node projections  q/k/v/skip = [x|feats] @ {Wq,Wk,Wv,Wskip} + bias
// Block = 32 nodes (two 16-row WMMA tiles); 8 waves, wave w owns output cols
// [16w,16w+16) of all four matrices. Register blocking: each B fragment
// (L2-resident weights) feeds 2 WMMAs, each A fragment feeds 4 matrices.
// ---------------------------------------------------------------------------
__global__ __launch_bounds__(256) void node_proj_kernel(
    const float* __restrict__ x, const float* __restrict__ feats,
    const float* __restrict__ Wq, const float* __restrict__ bq,
    const float* __restrict__ Wk, const float* __restrict__ bk,
    const float* __restrict__ Wv, const float* __restrict__ bv,
    const float* __restrict__ Wskip, const float* __restrict__ bskip,
    float* __restrict__ qb, float* __restrict__ kb, float* __restrict__ vb,
    float* __restrict__ outskip)
{
    __shared__ float hA[32][IN_CH + 1];   // +1 pad: kill LDS bank conflicts
    const int n0  = blockIdx.x * 32;
    const int tid = threadIdx.x;

    for (int i = tid; i < 32 * IN_CH; i += 256) {
        const int r = i >> 7, c = i & 127;
        const int n = n0 + r;
        hA[r][c] = (c < 64) ? x[(size_t)n * 64 + c]
                            : feats[(size_t)n * 64 + (c - 64)];
    }
    __syncthreads();

    const int lane  = tid & 31;
    const int row16 = lane & 15;
    const int hi    = lane >> 4;          // K sub-block 0/1 within A/B frags
    const int col   = (tid >> 5) * 16 + row16;

    v8f aq0 = {}; v8f aq1 = {};
    v8f ak0 = {}; v8f ak1 = {};
    v8f av0 = {}; v8f av1 = {};
    v8f as0 = {}; v8f as1 = {};
    for (int kk = 0; kk < IN_CH; kk += 4) {
        const int kf = kk + hi * 2;
        v2f a0; a0.x = hA[row16][kf];      a0.y = hA[row16][kf + 1];
        v2f a1; a1.x = hA[16 + row16][kf]; a1.y = hA[16 + row16][kf + 1];
        v2f b;
        b.x = Wq[kf * OUT_CH + col];    b.y = Wq[(kf + 1) * OUT_CH + col];
        aq0 = WMMA_F32(a0, b, aq0);     aq1 = WMMA_F32(a1, b, aq1);
        b.x = Wk[kf * OUT_CH + col];    b.y = Wk[(kf + 1) * OUT_CH + col];
        ak0 = WMMA_F32(a0, b, ak0);     ak1 = WMMA_F32(a1, b, ak1);
        b.x = Wv[kf * OUT_CH + col];    b.y = Wv[(kf + 1) * OUT_CH + col];
        av0 = WMMA_F32(a0, b, av0);     av1 = WMMA_F32(a1, b, av1);
        b.x = Wskip[kf * OUT_CH + col]; b.y = Wskip[(kf + 1) * OUT_CH + col];
        as0 = WMMA_F32(a0, b, as0);     as1 = WMMA_F32(a1, b, as1);
    }

    const float cq = bq[col], ck = bk[col], cv = bv[col], cs = bskip[col];
    #pragma unroll
    for (int i = 0; i < 8; ++i) {
        const int r = i + hi * 8;                      // C/D layout row
        const size_t o0 = (size_t)(n0 + r) * OUT_CH + col;
        const size_t o1 = (size_t)(n0 + 16 + r) * OUT_CH + col;
        qb[o0]      = aq0[i] + cq;  qb[o1]      = aq1[i] + cq;
        kb[o0]      = ak0[i] + ck;  kb[o1]      = ak1[i] + ck;
        vb[o0]      = av0[i] + cv;  vb[o1]      = av1[i] + cv;
        outskip[o0] = as0[i] + cs;  outskip[o1] = as1[i] + cs; // skip -> d_out
    }
}

// ---------------------------------------------------------------------------
// Kernel 2: e = [cos(rel_t*w + b) | msg] @ We   for a 32-edge tile per block.
// Edge-attr tile materialized in LDS only (never in HBM); 2 row tiles/wave.
// ---------------------------------------------------------------------------
__global__ __launch_bounds__(256) void edge_gemm_kernel(
    const int* __restrict__ ei,             // [2, NED]; row0=src, row1=dst
    const float* __restrict__ last_update,
    const float* __restrict__ t,
    const float* __restrict__ msg,
    const float* __restrict__ time_w, const float* __restrict__ time_b,
    const float* __restrict__ We,
    float* __restrict__ ebuf)
{
    __shared__ float attr[32][IN_CH + 1];
    __shared__ float relt[32];
    const int e0  = blockIdx.x * 32;
    const int tid = threadIdx.x;

    if (tid < 32) {
        const int s = ei[e0 + tid];
        relt[tid] = last_update[s] - t[e0 + tid];
    }
    __syncthreads();

    for (int i = tid; i < 32 * IN_CH; i += 256) {
        const int r = i >> 7, c = i & 127;
        float v;
        if (c < TIME_DIM) v = cosf(relt[r] * time_w[c] + time_b[c]);
        else              v = msg[(size_t)(e0 + r) * MSG_DIM + (c - TIME_DIM)];
        attr[r][c] = v;
    }
    __syncthreads();

    const int lane  = tid & 31;
    const int row16 = lane & 15;
    const int hi    = lane >> 4;
    const int col   = (tid >> 5) * 16 + row16;

    v8f acc0 = {}; v8f acc1 = {};
    for (int kk = 0; kk < IN_CH; kk += 4) {
        const int kf = kk + hi * 2;
        v2f a0; a0.x = attr[row16][kf];      a0.y = attr[row16][kf + 1];
        v2f a1; a1.x = attr[16 + row16][kf]; a1.y = attr[16 + row16][kf + 1];
        v2f b;  b.x  = We[kf * OUT_CH + col]; b.y = We[(kf + 1) * OUT_CH + col];
        acc0 = WMMA_F32(a0, b, acc0);
        acc1 = WMMA_F32(a1, b, acc1);
    }

    #pragma unroll
    for (int i = 0; i < 8; ++i) {
        const int r = i + hi * 8;
        ebuf[(size_t)(e0 + r) * OUT_CH + col]      = acc0[i];
        ebuf[(size_t)(e0 + 16 + r) * OUT_CH + col] = acc1[i];
    }
}

// Order-preserving float <-> uint key (for atomic segment-max).
__device__ __forceinline__ unsigned int f2key(float f) {
    unsigned int b = __float_as_uint(f);
    return (b & 0x80000000u) ? ~b : (b | 0x80000000u);
}
__device__ __forceinline__ float key2f(unsigned int k) {
    return __uint_as_float((k & 0x80000000u) ? (k ^ 0x80000000u) : ~k);
}

// ---------------------------------------------------------------------------
// Kernel 3: alpha[e,h] = dot(q[dst,h], k[src,h]+e[e,h]) / 4 ; segment max.
// ---------------------------------------------------------------------------
__global__ __launch_bounds__(256) void alpha_kernel(
    const int* __restrict__ ei, const float* __restrict__ q,
    const float* __restrict__ k, const float* __restrict__ ebuf,
    float* __restrict__ alpha, unsigned int* __restrict__ menc)
{
    const int idx = blockIdx.x * 256 + threadIdx.x;          // edge*8 + head
    if (idx >= NED * HEADS) return;
    const int e = idx >> 3, h = idx & 7;
    const int s = ei[e], d = ei[NED + e];

    const float4* qp = (const float4*)(q    + (size_t)d * OUT_CH + h * HEAD_DIM);
    const float4* kp = (const float4*)(k    + (size_t)s * OUT_CH + h * HEAD_DIM);
    const float4* ep = (const float4*)(ebuf + (size_t)e * OUT_CH + h * HEAD_DIM);

    float a = 0.f;
    #pragma unroll
    for (int i = 0; i < 4; ++i) {
        const float4 qv = qp[i], kv = kp[i], ev = ep[i];
        a += qv.x * (kv.x + ev.x) + qv.y * (kv.y + ev.y)
           + qv.z * (kv.z + ev.z) + qv.w * (kv.w + ev.w);
    }
    a *= 0.25f;                                              // 1/sqrt(16)
    alpha[idx] = a;
    atomicMax(&menc[(size_t)d * HEADS + h], f2key(a));
}

// ---------------------------------------------------------------------------
// Kernel 4: ex = exp(alpha - m[dst]); den += ex; num += ex*(v[src]+e)
// ---------------------------------------------------------------------------
__global__ __launch_bounds__(256) void accum_kernel(
    const int* __restrict__ ei, const float* __restrict__ v,
    const float* __restrict__ ebuf, const float* __restrict__ alpha,
    const unsigned int* __restrict__ menc,
    float* __restrict__ den, float* __restrict__ num)
{
    const int idx = blockIdx.x * 256 + threadIdx.x;
    if (idx >= NED * HEADS) return;
    const int e = idx >> 3, h = idx & 7;
    const int s = ei[e], d = ei[NED + e];

    const float m  = key2f(menc[(size_t)d * HEADS + h]);
    const float ex = expf(alpha[idx] - m);
    atomicAdd(&den[(size_t)d * HEADS + h], ex);

    const float4* vp = (const float4*)(v    + (size_t)s * OUT_CH + h * HEAD_DIM);
    const float4* ep = (const float4*)(ebuf + (size_t)e * OUT_CH + h * HEAD_DIM);
    float*        np = num + (size_t)d * OUT_CH + h * HEAD_DIM;
    #pragma unroll
    for (int i = 0; i < 4; ++i) {
        const float4 vv = vp[i], ev = ep[i];
        atomicAdd(&np[i * 4 + 0], ex * (vv.x + ev.x));
        atomicAdd(&np[i * 4 + 1], ex * (vv.y + ev.y));
        atomicAdd(&np[i * 4 + 2], ex * (vv.z + ev.z));
        atomicAdd(&np[i * 4 + 3], ex * (vv.w + ev.w));
    }
}

// ---------------------------------------------------------------------------
// Kernel 5: out = num / (den + 1e-16) + skip   (skip already in d_out)
// ---------------------------------------------------------------------------
__global__ __launch_bounds__(256) void finalize_kernel(
    const float* __restrict__ num, const float* __restrict__ den,
    float* __restrict__ out)
{
    const int i = blockIdx.x * 256 + threadIdx.x;
    if (i >= NND * OUT_CH) return;
    const int n = i >> 7, h = (i & 127) >> 4;
    out[i] = num[i] / (den[(size_t)n * HEADS + h] + 1e-16f) + out[i];
}

// ---------------------------------------------------------------------------
extern "C" void kernel_launch(void* const* d_in, const int* in_sizes, int n_in,
                              void* d_out, int out_size, void* d_ws, size_t ws_size,
                              hipStream_t stream)
{
    const float* feats       = (const float*)d_in[0];
    const float* x           = (const float*)d_in[1];
    const float* last_update = (const float*)d_in[2];
    const int*   edge_index  = (const int*)  d_in[3];
    const float* t           = (const float*)d_in[4];
    const float* msg         = (const float*)d_in[5];
    const float* time_w      = (const float*)d_in[6];
    const float* time_b      = (const float*)d_in[7];
    const float* Wq          = (const float*)d_in[8];
    const float* bq          = (const float*)d_in[9];
    const float* Wk          = (const float*)d_in[10];
    const float* bk          = (const float*)d_in[11];
    const float* Wv          = (const float*)d_in[12];
    const float* bv          = (const float*)d_in[13];
    const float* We          = (const float*)d_in[14];
    const float* Wskip       = (const float*)d_in[15];
    const float* bskip       = (const float*)d_in[16];
    float* out = (float*)d_out;

    // workspace layout (all chunks 256B-aligned by construction)
    char* ws = (char*)d_ws;
    float* qb    = (float*)ws; ws += (size_t)NND * OUT_CH * 4;   //  51.2 MB
    float* kb    = (float*)ws; ws += (size_t)NND * OUT_CH * 4;   //  51.2 MB
    float* vb    = (float*)ws; ws += (size_t)NND * OUT_CH * 4;   //  51.2 MB
    float* eb    = (float*)ws; ws += (size_t)NED * OUT_CH * 4;   // 256.0 MB
    float* alpha = (float*)ws; ws += (size_t)NED * HEADS  * 4;   //  16.0 MB
    unsigned int* menc = (unsigned int*)ws; ws += (size_t)NND * HEADS * 4;
    float* den   = (float*)ws; ws += (size_t)NND * HEADS  * 4;
    float* num   = (float*)ws; ws += (size_t)NND * OUT_CH * 4;

    // zero accumulators (menc/den/num are contiguous); key 0 == -inf sentinel
    hipMemsetAsync(menc, 0,
                   (size_t)NND * HEADS * 4 * 2 + (size_t)NND * OUT_CH * 4,
                   stream);

    node_proj_kernel<<<NND / 32, 256, 0, stream>>>(
        x, feats, Wq, bq, Wk, bk, Wv, bv, Wskip, bskip, qb, kb, vb, out);

    edge_gemm_kernel<<<NED / 32, 256, 0, stream>>>(
        edge_index, last_update, t, msg, time_w, time_b, We, eb);

    alpha_kernel<<<(NED * HEADS) / 256, 256, 0, stream>>>(
        edge_index, qb, kb, eb, alpha, menc);

    accum_kernel<<<(NED * HEADS) / 256, 256, 0, stream>>>(
        edge_index, vb, eb, alpha, menc, den, num);

    finalize_kernel<<<(NND * OUT_CH) / 256, 256, 0, stream>>>(num, den, out);
}